// GLoss_74500502716565
// MI455X (gfx1250) — compile-verified
//
#include <hip/hip_runtime.h>
#include <stdint.h>

// ---------------------------------------------------------------------------
// Problem geometry (from reference): output/target are [8,1,1024,1024] fp32.
// result = mse(grad8(t)-grad8(o)) + mse(o,t) + mse(grad8(pool2(t))-grad8(pool2(o)))
// grad8 is linear -> full-res term needs only d = t-o and its 8 neighbors.
// ---------------------------------------------------------------------------
constexpr int B  = 8;
constexpr int H  = 1024;
constexpr int W  = 1024;
constexpr int TW = 128;          // tile width  (image pixels)
constexpr int TH = 32;           // tile height
constexpr int RW = TW + 4;       // region width incl. 2-halo = 132
constexpr int RH = TH + 4;       // region height incl. 2-halo = 36
constexpr int SP = RW;           // LDS row pitch (floats)
constexpr int PW = TW / 2;       // pooled tile width  = 64
constexpr int PH = TH / 2;       // pooled tile height = 16
constexpr int PP = PW + 2;       // pooled pitch incl. 1-halo = 66
constexpr int NWG = (W / TW) * (H / TH) * B;   // 8*32*8 = 2048 workgroups

// mean denominators
constexpr float WG_G  = 1.0f / 67108864.0f;    // 8*8*1024*1024
constexpr float WG_P  = 1.0f / 8388608.0f;     // 8*1024*1024
constexpr float WG_GD = 1.0f / 16777216.0f;    // 8*8*512*512

// ---------------------------------------------------------------------------
// Async global->LDS path (CDNA5 GLOBAL_LOAD_ASYNC_TO_LDS_B32, ASYNCcnt).
// Probe: builtin exists; param 0 is address-space(1) int* ("__device__ int*").
// ---------------------------------------------------------------------------
#if defined(__has_builtin)
#if __has_builtin(__builtin_amdgcn_global_load_async_to_lds_b32)
#define ATH_HAVE_ASYNC_LDS 1
#endif
#endif
#ifndef ATH_HAVE_ASYNC_LDS
#define ATH_HAVE_ASYNC_LDS 0
#endif

typedef __attribute__((address_space(1))) int gbl_int_t;
typedef __attribute__((address_space(3))) int lds_int_t;

__device__ __forceinline__ void lds_copy_b32(const float* __restrict__ g, float* l) {
#if ATH_HAVE_ASYNC_LDS
  __builtin_amdgcn_global_load_async_to_lds_b32(
      (gbl_int_t*)(uintptr_t)g,
      (lds_int_t*)(uint32_t)(uintptr_t)l,
      /*imm offset*/ 0, /*cpol*/ 0);
#else
  *l = *g;                       // global_load_b32 + ds_store_b32 fallback
#endif
}

__device__ __forceinline__ void wait_async_lds() {
#if ATH_HAVE_ASYNC_LDS
#if __has_builtin(__builtin_amdgcn_s_wait_asynccnt)
  __builtin_amdgcn_s_wait_asynccnt(0);
#else
  asm volatile("s_wait_asynccnt 0" ::: "memory");
#endif
#endif
}

// ---------------------------------------------------------------------------
// Fused kernel: one workgroup per 128x32 tile of one image.
//   phase 1: stage o,t tile + 2-halo into LDS (zeros outside the image,
//            matching zero padding of the reference conv)
//   phase 2: full-res 8-direction gradient MSE + pixel MSE
//   phase 3: 2x2 maxpool of o and t separately -> pooled diff (with 1-pooled
//            halo) in LDS; pooled cells are 2x2-aligned so never straddle the
//            image border
//   phase 4: pooled 8-direction gradient MSE
//   phase 5: block reduction, one fp32 partial per workgroup (no atomics ->
//            deterministic under graph replay)
// ---------------------------------------------------------------------------
__global__ __launch_bounds__(256)
void GLoss_fused_kernel(const float* __restrict__ out_img,
                        const float* __restrict__ tgt_img,
                        float* __restrict__ partials) {
  __shared__ float so[RH * SP];        // output tile + halo
  __shared__ float st[RH * SP];        // target tile + halo
  __shared__ float spd[(PH + 2) * PP]; // pooled (t - o) with 1-pooled halo
  __shared__ float red[256];

  const int tid = threadIdx.x;
  const int x0 = blockIdx.x * TW;
  const int y0 = blockIdx.y * TH;
  const int img = blockIdx.z;
  const size_t base = (size_t)img * H * W;
  const float* __restrict__ op = out_img + base;
  const float* __restrict__ tp = tgt_img + base;

  // ---- phase 1: stage region [y0-2, y0+TH+2) x [x0-2, x0+TW+2) ----
  for (int i = tid; i < RH * RW; i += 256) {
    const int r = i / RW;
    const int c = i - r * RW;
    const int gy = y0 - 2 + r;
    const int gx = x0 - 2 + c;
    const int l = r * SP + c;
    if (gy >= 0 && gy < H && gx >= 0 && gx < W) {
      const size_t g = (size_t)gy * W + gx;
      lds_copy_b32(op + g, &so[l]);
      lds_copy_b32(tp + g, &st[l]);
    } else {
      so[l] = 0.0f;                    // zero padding
      st[l] = 0.0f;
    }
  }
  wait_async_lds();
  __syncthreads();

  float g_acc = 0.0f, p_acc = 0.0f, gd_acc = 0.0f;

  // ---- phase 2: full-res stencil. center at region (r+2, c+2) ----
  for (int i = tid; i < TH * TW; i += 256) {
    const int r = i >> 7;              // / TW (TW = 128)
    const int c = i & (TW - 1);
    const int l = (r + 2) * SP + (c + 2);
    const float dc = st[l] - so[l];
    p_acc += dc * dc;
#pragma unroll
    for (int dy = -1; dy <= 1; ++dy) {
#pragma unroll
      for (int dx = -1; dx <= 1; ++dx) {
        if (dy == 0 && dx == 0) continue;
        const int n = l + dy * SP + dx;
        const float dn = st[n] - so[n];
        const float e = dc - dn;
        g_acc += e * e;
      }
    }
  }

  // ---- phase 3: pooled diff with 1-pooled halo ----
  // spd(pr,pc) covers region rows 2*pr..2*pr+1, cols 2*pc..2*pc+1
  for (int i = tid; i < (PH + 2) * PP; i += 256) {
    const int pr = i / PP;
    const int pc = i - pr * PP;
    const int l = (2 * pr) * SP + 2 * pc;
    const float po = fmaxf(fmaxf(so[l], so[l + 1]), fmaxf(so[l + SP], so[l + SP + 1]));
    const float pt = fmaxf(fmaxf(st[l], st[l + 1]), fmaxf(st[l + SP], st[l + SP + 1]));
    spd[pr * PP + pc] = pt - po;       // fully-OOB pooled cells give 0 - 0 = 0
  }
  __syncthreads();

  // ---- phase 4: pooled 8-direction stencil ----
  for (int i = tid; i < PH * PW; i += 256) {
    const int pr = i >> 6;             // / PW (PW = 64)
    const int pc = i & (PW - 1);
    const int l = (pr + 1) * PP + (pc + 1);
    const float dc = spd[l];
#pragma unroll
    for (int dy = -1; dy <= 1; ++dy) {
#pragma unroll
      for (int dx = -1; dx <= 1; ++dx) {
        if (dy == 0 && dx == 0) continue;
        const float e = dc - spd[l + dy * PP + dx];
        gd_acc += e * e;
      }
    }
  }

  // ---- phase 5: weighted block reduction ----
  float part = g_acc * WG_G + p_acc * WG_P + gd_acc * WG_GD;
  red[tid] = part;
  __syncthreads();
#pragma unroll
  for (int s = 128; s > 0; s >>= 1) {
    if (tid < s) red[tid] += red[tid + s];
    __syncthreads();
  }
  if (tid == 0) {
    const int wg = (blockIdx.z * gridDim.y + blockIdx.y) * gridDim.x + blockIdx.x;
    partials[wg] = red[0];
  }
}

// Deterministic final reduction of the 2048 per-workgroup partials.
__global__ __launch_bounds__(256)
void GLoss_reduce_kernel(const float* __restrict__ partials, float* __restrict__ out) {
  __shared__ float s[256];
  const int tid = threadIdx.x;
  float a = 0.0f;
  for (int i = tid; i < NWG; i += 256) a += partials[i];
  s[tid] = a;
  __syncthreads();
  if (tid == 0) {
    float t = 0.0f;
#pragma unroll 8
    for (int i = 0; i < 256; ++i) t += s[i];
    out[0] = t;
  }
}

extern "C" void kernel_launch(void* const* d_in, const int* in_sizes, int n_in,
                              void* d_out, int out_size, void* d_ws, size_t ws_size,
                              hipStream_t stream) {
  (void)in_sizes; (void)n_in; (void)out_size; (void)ws_size;
  const float* out_img = (const float*)d_in[0];
  const float* tgt_img = (const float*)d_in[1];
  float* partials = (float*)d_ws;      // NWG floats = 8 KB of scratch

  dim3 grid(W / TW, H / TH, B);        // 8 x 32 x 8 = 2048 workgroups
  GLoss_fused_kernel<<<grid, 256, 0, stream>>>(out_img, tgt_img, partials);
  GLoss_reduce_kernel<<<1, 256, 0, stream>>>(partials, (float*)d_out);
}